// MessagePassingLayer_13073880449416
// MI455X (gfx1250) — compile-verified
//
#include <hip/hip_runtime.h>
#include <hip/hip_bf16.h>

typedef __attribute__((ext_vector_type(2))) float v2f;
typedef __attribute__((ext_vector_type(4))) float v4f;
typedef __attribute__((ext_vector_type(8))) float v8f;

#define DD    128      // hidden dim
#define TWOD  256      // 2*D
#define MT    32       // rows (edges/nodes) per wave tile
#define AP2   132      // A-tile row stride (128 data + 4 pad) -> conflict-free col reads
#define WP    288      // pair-packed weight row stride: 256 data + 32 pad (bank shift 32)

#define WMMA_F32(a, b, c) __builtin_amdgcn_wmma_f32_16x16x4_f32( \
        false, (a), false, (b), (short)0, (c), false, false)

// Repack W[K,128] (row-major) into pair-packed LDS layout:
//   sWp[(k>>1)*WP + 2*n + (k&1)] = W[k][n]
// so a B fragment (W[k][n], W[k+1][n]) is one aligned ds_load_b64.
__device__ __forceinline__ void stage_packed(const float* __restrict__ W,
                                             float* __restrict__ sWp,
                                             int K, int tid, int nthreads)
{
    const int tasks = (K >> 1) * 64;           // k-pairs * n-pairs
    for (int i = tid; i < tasks; i += nthreads) {
        const int kp = i >> 6;                 // k pair index
        const int np = i & 63;                 // n pair index (n = 2*np)
        const v2f g0 = *(const v2f*)&W[(2 * kp)     * DD + 2 * np];
        const v2f g1 = *(const v2f*)&W[(2 * kp + 1) * DD + 2 * np];
        v4f o;
        o.x = g0.x; o.y = g1.x; o.z = g0.y; o.w = g1.y;
        *(v4f*)&sWp[kp * WP + 4 * np] = o;     // 16B aligned
    }
}

// one K-half (128 wide) of a 32-row GEMM: acc[2][8] += A(32x128) @ B(128x128)
// bRowBase = pair-row offset into packed weights (0 or 64).
__device__ __forceinline__ void gemm_half(const float* __restrict__ sAw,
                                          const float* __restrict__ sWp,
                                          int bRowBase, int l16, int lhi,
                                          v8f acc[2][8])
{
    for (int k0 = 0; k0 < DD; k0 += 4) {
        const v2f a0 = *(const v2f*)&sAw[l16 * AP2 + k0 + 2 * lhi];
        const v2f a1 = *(const v2f*)&sAw[(16 + l16) * AP2 + k0 + 2 * lhi];
        const float* bp = &sWp[(bRowBase + (k0 >> 1) + lhi) * WP + l16 * 2];
#pragma unroll
        for (int t = 0; t < 8; ++t) {
            const v2f b = *(const v2f*)&bp[t * 32];
            acc[0][t] = WMMA_F32(a0, b, acc[0][t]);
            acc[1][t] = WMMA_F32(a1, b, acc[1][t]);
        }
    }
}

// ---------------------------------------------------------------------------
// Kernel 1: messages = silu(concat(nodes[src], edge_feat) @ mw1 + mb1)
//           scatter-add into agg[dst].  One wave = one 32-edge tile.
// ---------------------------------------------------------------------------
__launch_bounds__(256, 1)
__global__ void msg_kernel(const float* __restrict__ nodes,
                           const int*   __restrict__ edge_index,   // [2,E]
                           const float* __restrict__ edge_feat,    // [E,128]
                           const float* __restrict__ mw1,          // [256,128]
                           const float* __restrict__ mb1,          // [128]
                           float*       __restrict__ agg,          // [N,128]
                           int E, int iters)
{
    __shared__ __align__(16) float sW[(TWOD / 2) * WP];   // 144 KB, pair-packed
    __shared__ __align__(16) float sA[8][MT * AP2];       // 132 KB (16.5 KB/wave)

    const int tid  = threadIdx.x;
    const int w    = tid >> 5;
    const int lane = tid & 31;
    const int l16  = lane & 15;
    const int lhi  = lane >> 4;

    stage_packed(mw1, sW, TWOD, tid, 256);
    __syncthreads();

    const int* srcIdx = edge_index;
    const int* dstIdx = edge_index + E;
    const int totalWaves = gridDim.x * 8;
    const int nTiles = E >> 5;
    float* sAw = sA[w];

    for (int it = 0; it < iters; ++it) {
        const int tile = (blockIdx.x * 8 + w) + it * totalWaves;  // wave-uniform
        const bool active = tile < nTiles;
        const int e0 = tile << 5;

        // phase 0: gather node features (K-half 0 of the virtual concat)
        if (active) {
            for (int t = lane; t < MT * 32; t += 32) {
                const int r  = t >> 5;     // row (edge in tile)
                const int c4 = t & 31;     // float4 column
                const int s  = srcIdx[e0 + r];
                *(v4f*)&sAw[r * AP2 + c4 * 4] =
                    *(const v4f*)&nodes[(size_t)s * DD + c4 * 4];
            }
        }
        __syncthreads();

        v8f acc[2][8] = {};
        if (active)
            gemm_half(sAw, sW, 0, l16, lhi, acc);
        __syncthreads();   // all lanes done reading before restage

        // phase 1: edge features (K-half 1), streaming -> non-temporal
        if (active) {
            for (int t = lane; t < MT * 32; t += 32) {
                const int r  = t >> 5;
                const int c4 = t & 31;
                const v4f v = __builtin_nontemporal_load(
                    (const v4f*)&edge_feat[(size_t)(e0 + r) * DD + c4 * 4]);
                *(v4f*)&sAw[r * AP2 + c4 * 4] = v;
            }
        }
        __syncthreads();

        if (active) {
            gemm_half(sAw, sW, 64, l16, lhi, acc);

            // bias + SiLU + scatter-add.  C layout: (m=16s+j+8*lhi, n=16t+l16)
#pragma unroll
            for (int s = 0; s < 2; ++s) {
#pragma unroll
                for (int j = 0; j < 8; ++j) {
                    const int m = s * 16 + j + 8 * lhi;
                    const int d = dstIdx[e0 + m];
                    float* aggRow = agg + (size_t)d * DD;
#pragma unroll
                    for (int t = 0; t < 8; ++t) {
                        const int n = t * 16 + l16;
                        const float x = acc[s][t][j] + mb1[n];
                        const float v = x / (1.f + __expf(-x));
                        unsafeAtomicAdd(&aggRow[n], v);
                    }
                }
            }
        }
        __syncthreads();
    }
}

// ---------------------------------------------------------------------------
// Kernel 2: out = nodes + silu(concat(nodes, agg) @ uw1 + ub1) @ uw2 + ub2
// One wave = one 32-node tile. 4 waves / block.
// ---------------------------------------------------------------------------
__launch_bounds__(128, 1)
__global__ void upd_kernel(const float* __restrict__ nodes,
                           const float* __restrict__ agg,
                           const float* __restrict__ uw1,   // [256,128]
                           const float* __restrict__ ub1,   // [128]
                           const float* __restrict__ uw2,   // [128,128]
                           const float* __restrict__ ub2,   // [128]
                           float*       __restrict__ out,   // [N,128]
                           int N, int iters)
{
    __shared__ __align__(16) float sW1[(TWOD / 2) * WP];   // 144 KB
    __shared__ __align__(16) float sW2[(DD / 2) * WP];     //  72 KB
    __shared__ __align__(16) float sA[4][MT * AP2];        //  66 KB

    const int tid  = threadIdx.x;
    const int w    = tid >> 5;
    const int lane = tid & 31;
    const int l16  = lane & 15;
    const int lhi  = lane >> 4;

    stage_packed(uw1, sW1, TWOD, tid, 128);
    stage_packed(uw2, sW2, DD,   tid, 128);
    __syncthreads();

    const int totalWaves = gridDim.x * 4;
    const int nTiles = N >> 5;
    float* sAw = sA[w];

    for (int it = 0; it < iters; ++it) {
        const int tile = (blockIdx.x * 4 + w) + it * totalWaves;  // wave-uniform
        const bool active = tile < nTiles;
        const int i0 = tile << 5;

        // phase 0: node rows (K-half 0)
        if (active) {
            for (int t = lane; t < MT * 32; t += 32) {
                const int r  = t >> 5;
                const int c4 = t & 31;
                *(v4f*)&sAw[r * AP2 + c4 * 4] =
                    *(const v4f*)&nodes[(size_t)(i0 + r) * DD + c4 * 4];
            }
        }
        __syncthreads();

        v8f h[2][8] = {};
        if (active)
            gemm_half(sAw, sW1, 0, l16, lhi, h);
        __syncthreads();

        // phase 1: aggregated rows (K-half 1)
        if (active) {
            for (int t = lane; t < MT * 32; t += 32) {
                const int r  = t >> 5;
                const int c4 = t & 31;
                *(v4f*)&sAw[r * AP2 + c4 * 4] =
                    *(const v4f*)&agg[(size_t)(i0 + r) * DD + c4 * 4];
            }
        }
        __syncthreads();

        if (active)
            gemm_half(sAw, sW1, 64, l16, lhi, h);
        __syncthreads();   // done reading tile before hidden overwrite

        // bias + SiLU, write 32x128 hidden tile to LDS
        if (active) {
#pragma unroll
            for (int s = 0; s < 2; ++s) {
#pragma unroll
                for (int j = 0; j < 8; ++j) {
                    const int m = s * 16 + j + 8 * lhi;
#pragma unroll
                    for (int t = 0; t < 8; ++t) {
                        const int n = t * 16 + l16;
                        const float x = h[s][t][j] + ub1[n];
                        sAw[m * AP2 + n] = x / (1.f + __expf(-x));
                    }
                }
            }
        }
        __syncthreads();   // hidden tile visible to all lanes

        if (active) {
            v8f acc[2][8] = {};
            gemm_half(sAw, sW2, 0, l16, lhi, acc);

            // residual + bias2, streaming store
#pragma unroll
            for (int s = 0; s < 2; ++s) {
#pragma unroll
                for (int j = 0; j < 8; ++j) {
                    const int m = s * 16 + j + 8 * lhi;
                    const size_t row = (size_t)(i0 + m) * DD;
#pragma unroll
                    for (int t = 0; t < 8; ++t) {
                        const int n = t * 16 + l16;
                        const float v = nodes[row + n] + acc[s][t][j] + ub2[n];
                        __builtin_nontemporal_store(v, &out[row + n]);
                    }
                }
            }
        }
        __syncthreads();
    }
}

// ---------------------------------------------------------------------------
extern "C" void kernel_launch(void* const* d_in, const int* in_sizes, int n_in,
                              void* d_out, int out_size, void* d_ws, size_t ws_size,
                              hipStream_t stream) {
    const float* nodes     = (const float*)d_in[0];
    const int*   edge_idx  = (const int*)  d_in[1];
    const float* edge_feat = (const float*)d_in[2];
    const float* mw1       = (const float*)d_in[3];
    const float* mb1       = (const float*)d_in[4];
    const float* uw1       = (const float*)d_in[5];
    const float* ub1       = (const float*)d_in[6];
    const float* uw2       = (const float*)d_in[7];
    const float* ub2       = (const float*)d_in[8];
    float*       out       = (float*)d_out;

    const int N = in_sizes[0] / DD;    // 40000
    const int E = in_sizes[1] / 2;     // 640000

    float* agg = (float*)d_ws;         // N*DD floats of scratch
    hipMemsetAsync(agg, 0, (size_t)N * DD * sizeof(float), stream);

    // Kernel 1: E/32 tiles, 8 waves/block, ~8 tiles per wave
    const int tilesA  = E >> 5;
    const int wavesA  = (tilesA + 7) / 8;
    const int blocksA = (wavesA + 7) / 8;
    const int itersA  = (tilesA + blocksA * 8 - 1) / (blocksA * 8);
    msg_kernel<<<blocksA, 256, 0, stream>>>(nodes, edge_idx, edge_feat,
                                            mw1, mb1, agg, E, itersA);

    // Kernel 2: N/32 tiles, 4 waves/block, ~2 tiles per wave
    const int tilesB  = N >> 5;
    const int wavesB  = (tilesB + 1) / 2;
    const int blocksB = (wavesB + 3) / 4;
    const int itersB  = (tilesB + blocksB * 4 - 1) / (blocksB * 4);
    upd_kernel<<<blocksB, 128, 0, stream>>>(nodes, agg, uw1, ub1, uw2, ub2,
                                            out, N, itersB);
}